// DetectionLoss_7000796692536
// MI455X (gfx1250) — compile-verified
//
#include <hip/hip_runtime.h>
#include <hip/hip_bf16.h>
#include <math.h>

// Problem constants (fixed by the reference setup_inputs()).
constexpr int B = 16, P = 8192, N = 128, S = 5;
constexpr int SPLIT = 8;          // P-range splits for the matching pass
constexpr int PP = P / SPLIT;     // 1024 preds per match block
constexpr int K1T = 128;          // threads per match block (one per target n)
constexpr int C = 16;             // chunks per batch for elementwise partials
constexpr int CP = P / C;         // 512 preds per chunk
constexpr float EPS = 1e-6f;
constexpr float IOU_THRESH = 0.5f;

// ---- CDNA5 async copy helpers -------------------------------------------
// global_load_async_to_lds_b128: per-lane 16B copy memory -> LDS, tracked by
// ASYNCcnt (gfx1250). LDS destination address is a 32-bit byte offset in a
// VGPR; generic pointers to __shared__ truncate to exactly that offset.
__device__ __forceinline__ void async_copy_b128(void* lds_dst, const void* gsrc) {
    uint32_t dst = (uint32_t)(uintptr_t)lds_dst;
    asm volatile("global_load_async_to_lds_b128 %0, %1, off"
                 :: "v"(dst), "v"(gsrc)
                 : "memory");
}
__device__ __forceinline__ void wait_async0() {
#if __has_builtin(__builtin_amdgcn_s_wait_asynccnt)
    __builtin_amdgcn_s_wait_asynccnt(0);
#else
    asm volatile("s_wait_asynccnt 0" ::: "memory");
#endif
}

__device__ __forceinline__ float bce_logits(float x, float t) {
    return fmaxf(x, 0.f) - x * t + log1pf(expf(-fabsf(x)));
}
__device__ __forceinline__ float smooth_l1(float a, float b) {
    float d = fabsf(a - b);
    return (d < 1.f) ? (0.5f * d * d) : (d - 0.5f);
}

// ---- K1: best-IoU matching over a P-range, LDS-staged via async copies ----
__global__ __launch_bounds__(K1T)
void k_match(const float* __restrict__ pred_boxes,
             const float* __restrict__ tgt_boxes,
             float* __restrict__ best_val,
             int*   __restrict__ best_idx) {
    __shared__ float4 s_pb[PP];   // 16 KB pred-box tile
    __shared__ float4 s_tb[N];    // 2 KB target boxes for this batch
    const int tid = threadIdx.x;                 // = target index n
    const int b  = blockIdx.x / SPLIT;
    const int sp = blockIdx.x % SPLIT;
    const int p0 = sp * PP;

    // stage target boxes (one b128 per lane covers all 128 targets)
    async_copy_b128(&s_tb[tid], tgt_boxes + (size_t)(b * N + tid) * 4);
    // stage 1024 pred boxes: 8 async b128 issues per lane
#pragma unroll
    for (int i = 0; i < PP / K1T; ++i)
        async_copy_b128(&s_pb[i * K1T + tid],
                        pred_boxes + (size_t)(b * P + p0 + i * K1T + tid) * 4);
    wait_async0();
    __syncthreads();

    const float4 tb = s_tb[tid];
    const float areaB = (tb.z - tb.x) * (tb.w - tb.y);
    float best = -1.f;
    int   bi   = 0;
    for (int p = 0; p < PP; ++p) {
        const float4 pb = s_pb[p];     // ds_load_b128, broadcast across lanes
        float areaA = (pb.z - pb.x) * (pb.w - pb.y);
        float lx = fmaxf(pb.x, tb.x), ly = fmaxf(pb.y, tb.y);
        float rx = fminf(pb.z, tb.z), ry = fminf(pb.w, tb.w);
        float w = fmaxf(rx - lx, 0.f), h = fmaxf(ry - ly, 0.f);
        float inter = w * h;
        float iou = inter / (areaA + areaB - inter);
        if (iou > best) { best = iou; bi = p0 + p; }   // strict > => first max
    }
    best_val[(b * SPLIT + sp) * N + tid] = best;
    best_idx[(b * SPLIT + sp) * N + tid] = bi;
}

// ---- K2: elementwise partial sums (regularizers + sum bce(x,0)) ----------
__global__ __launch_bounds__(256)
void k_elem(const float* __restrict__ scores,
            const float* __restrict__ boxes,
            const float* __restrict__ scales,
            const float* __restrict__ ctx,
            float* __restrict__ partials) {
    __shared__ float r[5][256];
    const int b = blockIdx.x / C, c = blockIdx.x % C, tid = threadIdx.x;
    float a0 = 0, a1 = 0, a2 = 0, a3 = 0, a4 = 0;
    for (int i = tid; i < CP; i += 256) {
        const size_t off = (size_t)b * P + (size_t)c * CP + i;
        float x = scores[off];
        a0 += fmaxf(x, 0.f) + log1pf(expf(-fabsf(x)));   // bce(x, 0)
        a4 += x * x;
        const float* bx = boxes + off * 4;
#pragma unroll
        for (int k = 0; k < 4; ++k) a1 += bx[k] * bx[k];
        const float* sc = scales + off * S;
#pragma unroll
        for (int k = 0; k < S; ++k) a2 += sc[k] * sc[k];
        float cx = ctx[off];
        a3 += cx * cx;
    }
    r[0][tid] = a0; r[1][tid] = a1; r[2][tid] = a2; r[3][tid] = a3; r[4][tid] = a4;
    __syncthreads();
    for (int st = 128; st > 0; st >>= 1) {
        if (tid < st)
#pragma unroll
            for (int j = 0; j < 5; ++j) r[j][tid] += r[j][tid + st];
        __syncthreads();
    }
    if (tid == 0)
#pragma unroll
        for (int j = 0; j < 5; ++j) partials[(size_t)(b * C + c) * 5 + j] = r[j][0];
}

// ---- K3: per-(b,n) split-reduce, gather matched preds, loss sums ---------
__global__ __launch_bounds__(N)
void k_losses(const float* __restrict__ pred_scores,
              const float* __restrict__ pred_boxes,
              const float* __restrict__ pred_scales,
              const float* __restrict__ pred_context,
              const float* __restrict__ tgt_boxes,
              const int*   __restrict__ tgt_scales,
              const float* __restrict__ tgt_context,
              const float* __restrict__ best_val,
              const int*   __restrict__ best_idx,
              float* __restrict__ perb) {
    __shared__ int   s_bi[N];
    __shared__ float s_red[5][N];
    const int b = blockIdx.x, n = threadIdx.x;

    // reduce over P-splits; splits scanned in ascending-p order so strict >
    // preserves the global first-occurrence argmax
    float best = -1.f; int bi = 0;
    for (int sp = 0; sp < SPLIT; ++sp) {
        float v = best_val[(b * SPLIT + sp) * N + n];
        int   i = best_idx[(b * SPLIT + sp) * N + n];
        if (v > best) { best = v; bi = i; }
    }
    const bool valid = best > IOU_THRESH;
    s_bi[n] = valid ? bi : -1;

    float boxl = 0, cel = 0, ctxl = 0, cntv = 0, confm = 0;
    if (valid) {
        cntv = 1.f;
        const float* pb = pred_boxes + (size_t)(b * P + bi) * 4;
        const float* tb = tgt_boxes + (size_t)(b * N + n) * 4;
#pragma unroll
        for (int k = 0; k < 4; ++k) boxl += smooth_l1(pb[k], tb[k]);
        const float* ps = pred_scales + (size_t)(b * P + bi) * S;
        float m = ps[0];
#pragma unroll
        for (int k = 1; k < S; ++k) m = fmaxf(m, ps[k]);
        float se = 0.f;
#pragma unroll
        for (int k = 0; k < S; ++k) se += expf(ps[k] - m);
        const int t = tgt_scales[b * N + n];
        cel = (m + logf(se)) - ps[t];                // -log_softmax[t]
        ctxl = bce_logits(pred_context[b * P + bi], tgt_context[b * N + n]);
    }
    __syncthreads();
    // deterministic dedup of matched slots: lowest n owning this pred wins;
    // the contribution (pred_scores[b, bi]) depends only on bi, so order-safe.
    if (valid) {
        bool first = true;
        for (int m2 = 0; m2 < n; ++m2)
            if (s_bi[m2] == bi) { first = false; break; }
        if (first) confm = pred_scores[(size_t)b * P + bi];
    }
    s_red[0][n] = boxl; s_red[1][n] = cel; s_red[2][n] = ctxl;
    s_red[3][n] = cntv; s_red[4][n] = confm;
    __syncthreads();
    for (int st = N / 2; st > 0; st >>= 1) {
        if (n < st)
#pragma unroll
            for (int j = 0; j < 5; ++j) s_red[j][n] += s_red[j][n + st];
        __syncthreads();
    }
    if (n == 0)
#pragma unroll
        for (int j = 0; j < 5; ++j) perb[b * 5 + j] = s_red[j][0];
}

// ---- K4: combine per-batch sums into the 5 outputs -----------------------
__global__ __launch_bounds__(32)
void k_final(const float* __restrict__ perb,
             const float* __restrict__ partials,
             float* __restrict__ out) {
    __shared__ float s_l[B][4];
    const int b = threadIdx.x;
    if (b < B) {
        float bce0 = 0, ssb = 0, sss = 0, ssc = 0, sssc = 0;
        for (int c = 0; c < C; ++c) {
            const float* pp = partials + (size_t)(b * C + c) * 5;
            bce0 += pp[0]; ssb += pp[1]; sss += pp[2]; ssc += pp[3]; sssc += pp[4];
        }
        const float box_s = perb[b * 5 + 0], ce_s = perb[b * 5 + 1];
        const float ctx_s = perb[b * 5 + 2], cnt = perb[b * 5 + 3];
        const float confm = perb[b * 5 + 4];
        const bool has = cnt > 0.f;
        const float denom = fmaxf(cnt, 1.f);
        const float box_l   = box_s / (denom * 4.f);
        const float scale_l = ce_s / denom;
        const float ctx_l   = ctx_s / denom;
        const float conf_l  = (bce0 - confm) / (float)P;   // bce(x,1)-bce(x,0) = -x
        const float box_reg   = EPS * ssb  / (float)(P * 4);
        const float scale_reg = EPS * sss  / (float)(P * S);
        const float ctx_reg   = EPS * ssc  / (float)P;
        const float conf_reg  = EPS * sssc / (float)P;
        s_l[b][0] = has ? box_l   : box_reg;
        s_l[b][1] = has ? scale_l : scale_reg;
        s_l[b][2] = has ? ctx_l   : ctx_reg;
        s_l[b][3] = has ? conf_l  : conf_reg;
    }
    __syncthreads();
    if (b == 0) {
        float acc[4] = {0, 0, 0, 0};
        for (int i = 0; i < B; ++i)
#pragma unroll
            for (int j = 0; j < 4; ++j) acc[j] += s_l[i][j];
#pragma unroll
        for (int j = 0; j < 4; ++j) acc[j] /= (float)B;
        out[1] = acc[0]; out[2] = acc[1]; out[3] = acc[2]; out[4] = acc[3];
        out[0] = acc[0] + acc[1] + acc[2] + acc[3];
    }
}

extern "C" void kernel_launch(void* const* d_in, const int* in_sizes, int n_in,
                              void* d_out, int out_size, void* d_ws, size_t ws_size,
                              hipStream_t stream) {
    const float* pred_scores  = (const float*)d_in[0];
    const float* pred_boxes   = (const float*)d_in[1];
    const float* pred_scales  = (const float*)d_in[2];
    const float* pred_context = (const float*)d_in[3];
    const float* tgt_boxes    = (const float*)d_in[4];
    const int*   tgt_scales   = (const int*)d_in[5];
    const float* tgt_context  = (const float*)d_in[6];
    // d_in[7] (target_conf) is unused by the reference.

    // Workspace carve-up (~136 KB): every slot is written before it is read,
    // so no zero-init pass is required.
    float* best_val = (float*)d_ws;                         // B*SPLIT*N
    int*   best_idx = (int*)(best_val + B * SPLIT * N);     // B*SPLIT*N
    float* perb     = (float*)(best_idx + B * SPLIT * N);   // B*5
    float* partials = perb + B * 5;                         // B*C*5
    float* out = (float*)d_out;

    hipLaunchKernelGGL(k_match, dim3(B * SPLIT), dim3(K1T), 0, stream,
                       pred_boxes, tgt_boxes, best_val, best_idx);
    hipLaunchKernelGGL(k_elem, dim3(B * C), dim3(256), 0, stream,
                       pred_scores, pred_boxes, pred_scales, pred_context, partials);
    hipLaunchKernelGGL(k_losses, dim3(B), dim3(N), 0, stream,
                       pred_scores, pred_boxes, pred_scales, pred_context,
                       tgt_boxes, tgt_scales, tgt_context,
                       best_val, best_idx, perb);
    hipLaunchKernelGGL(k_final, dim3(1), dim3(32), 0, stream, perb, partials, out);
}